// CubECLayr_24532853194893
// MI455X (gfx1250) — compile-verified
//
#include <hip/hip_runtime.h>

typedef float v2f __attribute__((ext_vector_type(2)));
typedef float v8f __attribute__((ext_vector_type(8)));

#define ECC_STEPS 32
#define HBINS 33
#define HSTRIDE 34    // padded per-thread histogram row (spreads the 64 LDS banks)
#define STRIPS 4
#define SROWS 64      // rows owned per strip (plus 1 halo row except last strip)

__device__ __forceinline__ int ecc_bin(float f) {
    // bin = clip(ceil((f - 0) / (1/31)), 0, 32)
    int k = (int)__builtin_ceilf(f * 31.0f);
    k = k < 0 ? 0 : k;
    k = k > 32 ? 32 : k;
    return k;
}

__device__ __forceinline__ void async_row_load(const float* base, unsigned gbyte_off,
                                               unsigned lds_byte_addr) {
    // GVS mode: mem = SGPR64(base) + VGPR32(offset); dest = LDS byte address VGPR.
    asm volatile("global_load_async_to_lds_b32 %0, %1, %2"
                 :: "v"(lds_byte_addr), "v"(gbyte_off), "s"(base)
                 : "memory");
}

__global__ void ecc_zero_ws(int* __restrict__ ws, int n) {
    for (int i = threadIdx.x; i < n; i += 256) ws[i] = 0;
}

// One block per (image, strip). Streams SROWS(+halo) rows through an 8-slot LDS
// ring filled by async-to-LDS DMA, 6 rows ahead; one barrier per row.
__global__ __launch_bounds__(256) void ecc_hist_kernel(const float* __restrict__ x,
                                                       int* __restrict__ ws) {
    __shared__ int   hist[256 * HSTRIDE];   // per-thread private histograms (no contention)
    __shared__ float ring[8][256];          // async-filled row ring buffer

    const int tid    = threadIdx.x;
    const int strip  = blockIdx.x & (STRIPS - 1);
    const int bc     = blockIdx.x >> 2;
    const float* img = x + (size_t)bc * 256 * 256;
    const int r0     = strip * SROWS;
    const int rlimit = (strip < STRIPS - 1) ? (SROWS + 1) : SROWS;   // +1 halo row
    const int c      = tid;

    for (int i = tid; i < 256 * HSTRIDE; i += 256) hist[i] = 0;
    int* myhist = &hist[tid * HSTRIDE];
    // (zero-init is ordered before first atomic by the barrier inside iteration 0)

    // Prologue: issue async loads for rows 0..5 of this strip (rlimit >= 64 > 6).
#pragma unroll
    for (int r = 0; r < 6; ++r) {
        unsigned lds = (unsigned)(unsigned long long)(&ring[r & 7][c]);
        unsigned off = (unsigned)(((r0 + r) * 256 + c) * 4);
        async_row_load(img, off, lds);
    }

    float up = 0.0f, upright = 0.0f;
    for (int r = 0; r < rlimit; ++r) {
        // Row r was issued 6 iterations ago: <=5 outstanding leaves it complete.
        if (r + 6 < rlimit) asm volatile("s_wait_asynccnt 0x5" ::: "memory");
        else                asm volatile("s_wait_asynccnt 0x0" ::: "memory");
        __syncthreads();    // all 8 waves' columns of row r are in LDS

        float cur   = ring[r & 7][c];
        float right = (c < 255) ? ring[r & 7][c + 1] : 0.0f;

        if (r < SROWS) {    // own rows only (halo row contributes no vertex/h-edge)
            atomicAdd(&myhist[ecc_bin(cur)], 1);                          // vertex  +1
            if (c < 255)
                atomicAdd(&myhist[ecc_bin(fmaxf(cur, right))], -1);       // h-edge  -1
        }
        if (r >= 1) {       // cell pair (row r-1, row r): up/upright carried in regs
            atomicAdd(&myhist[ecc_bin(fmaxf(up, cur))], -1);              // v-edge  -1
            if (c < 255) {
                float m4 = fmaxf(fmaxf(cur, right), fmaxf(up, upright));
                atomicAdd(&myhist[ecc_bin(m4)], 1);                       // square  +1
            }
        }

        // Refill: row r+6 lands in slot (r+6)&7 == (r-2)&7; its old contents (row r-2)
        // were last read 2 barriers ago by every wave -> safe to overwrite.
        if (r + 6 < rlimit) {
            unsigned lds = (unsigned)(unsigned long long)(&ring[(r + 6) & 7][c]);
            unsigned off = (unsigned)(((r0 + r + 6) * 256 + c) * 4);
            async_row_load(img, off, lds);
        }
        up = cur; upright = right;
    }
    __syncthreads();

    // Reduce 256 private histograms and accumulate strip partial into global scratch.
    if (tid < HBINS) {
        int s = 0;
        for (int t = 0; t < 256; ++t) s += hist[t * HSTRIDE + tid];
        atomicAdd(&ws[bc * HBINS + tid], s);   // exact integer atomics: deterministic
    }
}

// One wave32 per image: exact f32 triangular-matmul prefix sum via chained
// V_WMMA_F32_16X16X4_F32 (integer-valued f32 -> exact). EXEC is all-ones.
__global__ __launch_bounds__(32) void ecc_cumsum_kernel(const int* __restrict__ ws,
                                                        float* __restrict__ out) {
    __shared__ float wsc[2][32][8];
    const int bc   = blockIdx.x;
    const int lane = threadIdx.x & 31;
    const int half = lane >> 4;
    const int m    = lane & 15;
    const int* h   = ws + bc * HBINS;

    // Layouts (ISA 7.12.2): A 16x4: M=lane&15, K = vgpr + 2*half
    //                       B 4x16: K = vgpr + 2*half, N = lane&15
    //                       C/D:    M = vgpr + 8*half, N = lane&15
    v8f dlo = {};
    v8f dhi = {};
#pragma unroll
    for (int chunk = 0; chunk < 8; ++chunk) {
        const int k0 = chunk * 4 + half * 2;
        const int k1 = k0 + 1;
        v2f b;   b.x   = (float)h[k0];                 b.y   = (float)h[k1];
        v2f alo; alo.x = (k0 <= m)      ? 1.0f : 0.0f; alo.y = (k1 <= m)      ? 1.0f : 0.0f;
        v2f ahi; ahi.x = (k0 <= m + 16) ? 1.0f : 0.0f; ahi.y = (k1 <= m + 16) ? 1.0f : 0.0f;
        dlo = __builtin_amdgcn_wmma_f32_16x16x4_f32(false, alo, false, b, (short)0, dlo, false, false);
        dhi = __builtin_amdgcn_wmma_f32_16x16x4_f32(false, ahi, false, b, (short)0, dhi, false, false);
    }

#pragma unroll
    for (int i = 0; i < 8; ++i) {
        wsc[0][lane][i] = dlo[i];
        wsc[1][lane][i] = dhi[i];
    }
    __syncthreads();

    const int j = lane;
    float cv;
    if (j < 16) {
        cv = wsc[0][(j >> 3) * 16][j & 7];        // D[m=j][n=0]
    } else {
        const int jj = j - 16;
        cv = wsc[1][(jj >> 3) * 16][jj & 7];      // D[m=j-16][n=0]
    }
    out[(size_t)bc * ECC_STEPS + j] = cv;
}

extern "C" void kernel_launch(void* const* d_in, const int* in_sizes, int n_in,
                              void* d_out, int out_size, void* d_ws, size_t ws_size,
                              hipStream_t stream) {
    (void)n_in; (void)out_size; (void)ws_size;
    const float* x   = (const float*)d_in[0];
    float*       out = (float*)d_out;
    int*         ws  = (int*)d_ws;
    const int nbc = in_sizes[0] / (256 * 256);   // 64*3 = 192 images

    ecc_zero_ws<<<1, 256, 0, stream>>>(ws, nbc * HBINS);
    ecc_hist_kernel<<<nbc * STRIPS, 256, 0, stream>>>(x, ws);
    ecc_cumsum_kernel<<<nbc, 32, 0, stream>>>(ws, out);
}